// GIN_64158221467749
// MI455X (gfx1250) — compile-verified
//
#include <hip/hip_runtime.h>

typedef __attribute__((ext_vector_type(2))) float v2f;
typedef __attribute__((ext_vector_type(8))) float v8f;

// ---------------- init: agg = x (float4 / b128 copies) ----------------
__global__ void copy4_kernel(const float4* __restrict__ in, float4* __restrict__ out, int n4) {
  int i = blockIdx.x * blockDim.x + threadIdx.x;
  if (i < n4) out[i] = in[i];
}

// ---------------- edge gather + scatter-add (L2-resident) ----------------
// thread t handles edge e = t/16, feature chunk of 4 floats.
// Hardware f32 atomics (non-returning -> STOREcnt tracked, L2 RMW units).
__global__ void edge_scatter_kernel(const float* __restrict__ x,
                                    const int* __restrict__ src,
                                    const int* __restrict__ dst,
                                    const float* __restrict__ ew,
                                    float* __restrict__ agg, int E) {
  int gid = blockIdx.x * blockDim.x + threadIdx.x;
  int e = gid >> 4;
  if (e >= E) return;
  int c = (gid & 15) << 2;
  int s = src[e];
  int d = dst[e];
  float w = ew[e];
  const float4 v = *(const float4*)(x + (size_t)s * 64 + c);
  unsigned long long o = (unsigned long long)(agg + (size_t)d * 64 + c);
  float a0 = v.x * w, a1 = v.y * w, a2 = v.z * w, a3 = v.w * w;
  asm volatile(
      "global_atomic_add_f32 %0, %1, off scope:SCOPE_DEV\n\t"
      "global_atomic_add_f32 %0, %2, off offset:4 scope:SCOPE_DEV\n\t"
      "global_atomic_add_f32 %0, %3, off offset:8 scope:SCOPE_DEV\n\t"
      "global_atomic_add_f32 %0, %4, off offset:12 scope:SCOPE_DEV"
      :: "v"(o), "v"(a0), "v"(a1), "v"(a2), "v"(a3)
      : "memory");
}

// ---------------- fused GIN MLP: relu(H @ W1 + b1) @ W2 + b2 ----------------
// One wave per 16-node tile, fp32 WMMA 16x16x4 (full f32 accuracy).
// A-frag (ISA 7.12.2, 32-bit A 16x4): lane m (0-15): K=0,1 ; lane m+16: K=2,3.
// C/D: lane l holds col n=l%16, VGPR j holds row m = j + 8*(l/16).
// Weights are cooperatively staged in LDS pre-swizzled into B-fragment layout:
// fragment f = 32 lanes x v2f contiguous -> one conflict-free ds_load_b64 per WMMA.
template <int MID, int WAVES, bool RELU_OUT>
__global__ __launch_bounds__(WAVES * 32)
void gin_mlp_kernel(const float* __restrict__ H,
                    const float* __restrict__ W1, const float* __restrict__ B1,
                    const float* __restrict__ W2, const float* __restrict__ B2,
                    float* __restrict__ OUT, int N) {
  constexpr int  NT1    = MID / 16;      // matmul1 output tiles
  constexpr int  F1     = 16 * NT1;      // W1 fragments (16 K-steps x NT1)
  constexpr bool STAGE2 = (MID == 64);   // W2 frags also fit in LDS for 64-wide layers
  constexpr int  F2     = STAGE2 ? (MID / 4) * 4 : 0;
  constexpr int  LDSW   = MID + 4;       // padded Y1 row stride (bank-conflict free re-read)

  __shared__ float wlds[(F1 + F2) * 64];        // fragment-layout weights (<=32KB)
  __shared__ float yst_all[WAVES * 16 * LDSW];  // Y1 staging

  const int lane  = threadIdx.x & 31;
  const int wave  = threadIdx.x >> 5;
  const int m0    = (blockIdx.x * WAVES + wave) * 16;
  float* yst = yst_all + wave * 16 * LDSW;
  const int m     = lane & 15;
  const int hi    = lane >> 4;    // 0 or 1
  const int khalf = hi << 1;      // 0 or 2

  // ---- cooperative staging of weight fragments into LDS ----
  for (int s = threadIdx.x; s < (F1 + F2) * 32; s += WAVES * 32) {
    const int f  = s >> 5;          // fragment id
    const int l  = s & 31;          // lane slot within fragment
    const int fn = l & 15;
    const int fk = (l >> 4) << 1;
    v2f val;
    bool isW1 = true;
    if constexpr (STAGE2) isW1 = (f < F1);
    if (isW1) {
      const int kk = f / NT1;
      const int nt = f - kk * NT1;
      const int kb = kk * 4 + fk;
      const int n  = nt * 16 + fn;
      val.x = W1[(size_t)kb * MID + n];
      val.y = W1[(size_t)(kb + 1) * MID + n];
    } else {
      const int f2 = f - F1;
      const int kk = f2 >> 2;
      const int nt = f2 & 3;
      const int kb = kk * 4 + fk;
      const int n  = nt * 16 + fn;
      val.x = W2[(size_t)kb * 64 + n];
      val.y = W2[(size_t)(kb + 1) * 64 + n];
    }
    *(v2f*)&wlds[(size_t)s * 2] = val;
  }
  __syncthreads();

  if (m0 < N) {
    v8f z = {};
    v8f acc1[NT1];
#pragma unroll
    for (int i = 0; i < NT1; ++i) acc1[i] = z;

    // A fragments of the 16x64 input tile: 16 K-steps of 4.
    const float* hrow = H + (size_t)(m0 + m) * 64;
    v2f afrag[16];
#pragma unroll
    for (int kk = 0; kk < 16; ++kk)
      afrag[kk] = *(const v2f*)(hrow + kk * 4 + khalf);

    // First matmul: 16x64 @ 64xMID, B frags from LDS (one ds_load_b64 each)
#pragma unroll
    for (int kk = 0; kk < 16; ++kk) {
#pragma unroll
      for (int nt = 0; nt < NT1; ++nt) {
        v2f b = *(const v2f*)&wlds[(size_t)((kk * NT1 + nt) * 32 + lane) * 2];
        acc1[nt] = __builtin_amdgcn_wmma_f32_16x16x4_f32(
            false, afrag[kk], false, b, (short)0, acc1[nt], false, false);
      }
    }

    // bias + ReLU, stage Y1 (16 x MID) row-major in LDS
#pragma unroll
    for (int nt = 0; nt < NT1; ++nt) {
      const int n = nt * 16 + m;
      const float bias = B1[n];
#pragma unroll
      for (int j = 0; j < 8; ++j) {
        float v = acc1[nt][j] + bias;
        v = v > 0.0f ? v : 0.0f;
        yst[(j + 8 * hi) * LDSW + n] = v;
      }
    }
  }
  __syncthreads();

  if (m0 < N) {
    v8f z = {};
    v8f acc2[4];
#pragma unroll
    for (int i = 0; i < 4; ++i) acc2[i] = z;

    // Second matmul: 16xMID @ MIDx64
#pragma unroll
    for (int kk = 0; kk < MID / 4; ++kk) {
      const int kb = kk * 4 + khalf;
      v2f a2 = *(const v2f*)(yst + m * LDSW + kb);
#pragma unroll
      for (int nt = 0; nt < 4; ++nt) {
        v2f b;
        if constexpr (STAGE2) {
          b = *(const v2f*)&wlds[(size_t)((F1 + kk * 4 + nt) * 32 + lane) * 2];
        } else {
          const float* w2r = W2 + (size_t)kb * 64 + m;
          b.x = w2r[nt * 16];
          b.y = w2r[64 + nt * 16];
        }
        acc2[nt] = __builtin_amdgcn_wmma_f32_16x16x4_f32(
            false, a2, false, b, (short)0, acc2[nt], false, false);
      }
    }

    float* obase = OUT + (size_t)m0 * 64;
#pragma unroll
    for (int nt = 0; nt < 4; ++nt) {
      const int n = nt * 16 + m;
      const float bias = B2[n];
#pragma unroll
      for (int j = 0; j < 8; ++j) {
        float v = acc2[nt][j] + bias;
        if (RELU_OUT) v = v > 0.0f ? v : 0.0f;
        obase[(size_t)(j + 8 * hi) * 64 + n] = v;
      }
    }
  }
}

extern "C" void kernel_launch(void* const* d_in, const int* in_sizes, int n_in,
                              void* d_out, int out_size, void* d_ws, size_t ws_size,
                              hipStream_t stream) {
  const float* x   = (const float*)d_in[0];
  const int*   ei  = (const int*)d_in[1];
  const float* ew  = (const float*)d_in[2];
  const float* w11 = (const float*)d_in[3];
  const float* b11 = (const float*)d_in[4];
  const float* w12 = (const float*)d_in[5];
  const float* b12 = (const float*)d_in[6];
  const float* w21 = (const float*)d_in[7];
  const float* b21 = (const float*)d_in[8];
  const float* w22 = (const float*)d_in[9];
  const float* b22 = (const float*)d_in[10];
  const float* w31 = (const float*)d_in[11];
  const float* b31 = (const float*)d_in[12];
  const float* w32 = (const float*)d_in[13];
  const float* b32 = (const float*)d_in[14];

  float* out = (float*)d_out;
  float* agg = (float*)d_ws;          // N*64 floats scratch

  const int N  = in_sizes[0] / 64;
  const int E  = in_sizes[2];
  const int n4 = N * 16;              // N*64/4 float4 elements
  const int tiles = (N + 15) / 16;

  dim3 blk(256);
  dim3 gcopy((n4 + 255) / 256);
  dim3 gedge((E * 16 + 255) / 256);
  const int* src = ei;
  const int* dst = ei + E;

  // ---- layer 1: 64 -> 64 -> 64, outer ReLU ----
  copy4_kernel<<<gcopy, blk, 0, stream>>>((const float4*)x, (float4*)agg, n4);
  edge_scatter_kernel<<<gedge, blk, 0, stream>>>(x, src, dst, ew, agg, E);
  gin_mlp_kernel<64, 4, true><<<(tiles + 3) / 4, dim3(128), 0, stream>>>(
      agg, w11, b11, w12, b12, out, N);

  // ---- layer 2: 64 -> 128 -> 64, outer ReLU ----
  copy4_kernel<<<gcopy, blk, 0, stream>>>((const float4*)out, (float4*)agg, n4);
  edge_scatter_kernel<<<gedge, blk, 0, stream>>>(out, src, dst, ew, agg, E);
  gin_mlp_kernel<128, 2, true><<<(tiles + 1) / 2, dim3(64), 0, stream>>>(
      agg, w21, b21, w22, b22, out, N);

  // ---- layer 3: 64 -> 64 -> 64, no outer ReLU ----
  copy4_kernel<<<gcopy, blk, 0, stream>>>((const float4*)out, (float4*)agg, n4);
  edge_scatter_kernel<<<gedge, blk, 0, stream>>>(out, src, dst, ew, agg, E);
  gin_mlp_kernel<64, 4, false><<<(tiles + 7) / 8, dim3(128), 0, stream>>>(
      agg, w31, b31, w32, b32, out, N);
}